// GCNII_13975823581435
// MI455X (gfx1250) — compile-verified
//
#include <hip/hip_runtime.h>
#include <math.h>

#define NN 100000
#define EE 1600000
#define FIN 256
#define HH 64
#define CC 64
#define LL 8

typedef float v2f __attribute__((ext_vector_type(2)));
typedef float v8f __attribute__((ext_vector_type(8)));
typedef unsigned int v4u __attribute__((ext_vector_type(4)));
typedef int v4i __attribute__((ext_vector_type(4)));
typedef int v8i __attribute__((ext_vector_type(8)));

__device__ __forceinline__ float elu1(float v) {
    return v > 0.0f ? v : (__expf(v) - 1.0f);
}

// ---- TDM: DMA a 1-D run of f32 from global into LDS (CDNA5 tensor data mover).
// D# per ISA ch.8: group0 = {count|flags, lds_addr, global_addr lo, global_addr hi|type=2},
// group1 = {data_size=4B, tensor_dim0=tile_dim0=nelem, tensor_dim1=tile_dim1=1, stride}.
// Issue from one wave, then S_WAIT_TENSORCNT 0 before the workgroup barrier.
__device__ __forceinline__ void tdm_load_w(const float* gsrc, float* ldst, int nelem) {
    unsigned long long ga = (unsigned long long)(const void*)gsrc;
    unsigned lofs = (unsigned)(size_t)(void*)ldst;   // flat shared addr low 32b == LDS byte offset
    v4u g0 = { 1u,                                                  // count=1, user descriptor
               lofs,                                                // lds_addr (bytes)
               (unsigned)ga,                                        // global_addr[31:0]
               ((unsigned)(ga >> 32) & 0x01FFFFFFu) | 0x80000000u };// global_addr[56:32] | type=2
    v8i g1 = { 0x00020000,                                          // wg_mask=0, data_size=2 (4B)
               (int)(((unsigned)nelem & 0xFFFFu) << 16),            // tensor_dim0[15:0]
               (int)((((unsigned)nelem >> 16) & 0xFFFFu) | (1u << 16)), // dim0[31:16] | tensor_dim1=1
               (int)(((unsigned)nelem & 0xFFFFu) << 16),            // tile_dim0
               1,                                                   // tile_dim1=1, tile_dim2=0
               nelem,                                               // tensor_dim0_stride[31:0]
               0, 0 };
    v4i gz = { 0, 0, 0, 0 };
#if __clang_major__ >= 23
    v8i gz8 = { 0, 0, 0, 0, 0, 0, 0, 0 };
    __builtin_amdgcn_tensor_load_to_lds(g0, g1, gz, gz, gz8, 0);
#else
    __builtin_amdgcn_tensor_load_to_lds(g0, g1, gz, gz, 0);
#endif
    __builtin_amdgcn_s_wait_tensorcnt(0);
}

// ---------------- degree / norm ----------------
__global__ void k_zero_deg(unsigned* __restrict__ deg) {
    int i = blockIdx.x * 256 + threadIdx.x;
    if (i < NN) deg[i] = 0u;
}

__global__ void k_count_deg(const int* __restrict__ dst, unsigned* __restrict__ deg) {
    int e = blockIdx.x * 256 + threadIdx.x;
    if (e < EE) atomicAdd(&deg[dst[e]], 1u);
}

__global__ void k_dinv(const unsigned* __restrict__ deg, float* __restrict__ dinv) {
    int i = blockIdx.x * 256 + threadIdx.x;
    if (i < NN) dinv[i] = rsqrtf((float)(deg[i] + 1u));  // +1 self loop, always > 0
}

// ---------------- input GEMM: h = x0 = elu(x @ w_in + b_in) ----------------
// Weights staged pair-interleaved: lw[(k/2)*128 + 2n + (k&1)] so each B fragment
// {W[k][c], W[k+1][c]} is one aligned ds_load_b64 into a WMMA-ready VGPR pair.
__global__ __launch_bounds__(256) void k_gemm_in(const float* __restrict__ x,
                                                 const float* __restrict__ w_in,
                                                 const float* __restrict__ b_in,
                                                 float* __restrict__ h,
                                                 float* __restrict__ x0) {
    __shared__ float lw[FIN * HH];  // 64 KB, swizzled
    for (int i = threadIdx.x; i < FIN * HH; i += 256) {
        int k = i >> 6, n = i & 63;
        lw[(k >> 1) * 128 + 2 * n + (k & 1)] = w_in[i];
    }
    __syncthreads();

    const int wave = threadIdx.x >> 5;
    const int lane = threadIdx.x & 31;
    const int rt = blockIdx.x * 8 + wave;          // 16-row tile index
    if (rt >= NN / 16) return;                     // wave-granular: EXEC stays all-ones
    const int r16  = lane & 15;                    // A: row / B,C: col
    const int half = lane >> 4;
    const int kp   = half << 1;                    // k offset within K=4 step

    const float* xrow = x + (size_t)(rt * 16 + r16) * FIN;
    v8f c0 = {}, c1 = {}, c2 = {}, c3 = {};
    for (int k0 = 0; k0 < FIN; k0 += 4) {
        v2f a = *(const v2f*)(xrow + k0 + kp);
        const float* wb = lw + ((k0 + kp) >> 1) * 128 + 2 * r16;
        v2f b0 = *(const v2f*)(wb + 0);
        v2f b1 = *(const v2f*)(wb + 32);
        v2f b2 = *(const v2f*)(wb + 64);
        v2f b3 = *(const v2f*)(wb + 96);
        c0 = __builtin_amdgcn_wmma_f32_16x16x4_f32(false, a, false, b0, (short)0, c0, false, false);
        c1 = __builtin_amdgcn_wmma_f32_16x16x4_f32(false, a, false, b1, (short)0, c1, false, false);
        c2 = __builtin_amdgcn_wmma_f32_16x16x4_f32(false, a, false, b2, (short)0, c2, false, false);
        c3 = __builtin_amdgcn_wmma_f32_16x16x4_f32(false, a, false, b3, (short)0, c3, false, false);
    }

    const float bi0 = b_in[r16], bi1 = b_in[16 + r16], bi2 = b_in[32 + r16], bi3 = b_in[48 + r16];
#pragma unroll
    for (int v = 0; v < 8; ++v) {
        const int row = rt * 16 + v + 8 * half;
        const size_t o = (size_t)row * HH + r16;
        float e0 = elu1(c0[v] + bi0); h[o +  0] = e0; x0[o +  0] = e0;
        float e1 = elu1(c1[v] + bi1); h[o + 16] = e1; x0[o + 16] = e1;
        float e2 = elu1(c2[v] + bi2); h[o + 32] = e2; x0[o + 32] = e2;
        float e3 = elu1(c3[v] + bi3); h[o + 48] = e3; x0[o + 48] = e3;
    }
}

// ---------------- SpMM: agg = dinv^2 * h (self loop), then scatter edges ----------------
__global__ void k_init_agg(const float* __restrict__ dinv, const float* __restrict__ h,
                           float* __restrict__ agg) {
    int idx = blockIdx.x * 256 + threadIdx.x;
    if (idx < NN * HH) {
        int i = idx >> 6;
        float w = dinv[i];
        agg[idx] = w * w * h[idx];
    }
}

__global__ void k_scatter(const int* __restrict__ src, const int* __restrict__ dst,
                          const float* __restrict__ dinv, const float* __restrict__ h,
                          float* __restrict__ agg) {
    long long idx = (long long)blockIdx.x * 256 + threadIdx.x;
    int e = (int)(idx >> 4);          // 16 threads per edge
    if (e < EE) {
        int fo = ((int)idx & 15) * 4; // 4 features per thread
        int s = src[e], d = dst[e];
        float w = dinv[s] * dinv[d];
        const float4 hv = *(const float4*)(h + (size_t)s * HH + fo);
        float* ap = agg + (size_t)d * HH + fo;
        atomicAdd(ap + 0, w * hv.x);
        atomicAdd(ap + 1, w * hv.y);
        atomicAdd(ap + 2, w * hv.z);
        atomicAdd(ap + 3, w * hv.w);
    }
}

// ---------------- layer GEMM (fused epilogue): s = .5*agg+.5*x0;
// out = (1-beta)*s + beta*(s@W); h = elu(out)+out.
// Weight tile DMA'd by the TDM into LDS, then swizzled pair-interleaved locally.
__global__ __launch_bounds__(256) void k_gemm_layer(const float* __restrict__ agg,
                                                    const float* __restrict__ x0,
                                                    const float* __restrict__ wl,
                                                    float beta,
                                                    float* __restrict__ h) {
    __shared__ float lwr[HH * HH];  // raw tile (TDM destination), 16 KB
    __shared__ float lw[HH * HH];   // pair-interleaved, 16 KB
    if (threadIdx.x < 32) tdm_load_w(wl, lwr, HH * HH);
    __syncthreads();
    for (int i = threadIdx.x; i < HH * HH; i += 256) {
        int k = i >> 6, n = i & 63;
        lw[(k >> 1) * 128 + 2 * n + (k & 1)] = lwr[i];
    }
    __syncthreads();

    const int wave = threadIdx.x >> 5;
    const int lane = threadIdx.x & 31;
    const int rt = blockIdx.x * 8 + wave;
    if (rt >= NN / 16) return;
    const int r16  = lane & 15;
    const int half = lane >> 4;
    const int kp   = half << 1;

    const float* arow = agg + (size_t)(rt * 16 + r16) * HH;
    const float* xrow = x0  + (size_t)(rt * 16 + r16) * HH;
    v8f c0 = {}, c1 = {}, c2 = {}, c3 = {};
    for (int k0 = 0; k0 < HH; k0 += 4) {
        v2f av = *(const v2f*)(arow + k0 + kp);
        v2f xv = *(const v2f*)(xrow + k0 + kp);
        v2f a = { 0.5f * (av.x + xv.x), 0.5f * (av.y + xv.y) };
        const float* wb = lw + ((k0 + kp) >> 1) * 128 + 2 * r16;
        v2f b0 = *(const v2f*)(wb + 0);
        v2f b1 = *(const v2f*)(wb + 32);
        v2f b2 = *(const v2f*)(wb + 64);
        v2f b3 = *(const v2f*)(wb + 96);
        c0 = __builtin_amdgcn_wmma_f32_16x16x4_f32(false, a, false, b0, (short)0, c0, false, false);
        c1 = __builtin_amdgcn_wmma_f32_16x16x4_f32(false, a, false, b1, (short)0, c1, false, false);
        c2 = __builtin_amdgcn_wmma_f32_16x16x4_f32(false, a, false, b2, (short)0, c2, false, false);
        c3 = __builtin_amdgcn_wmma_f32_16x16x4_f32(false, a, false, b3, (short)0, c3, false, false);
    }

    const float ob = 1.0f - beta;
#pragma unroll
    for (int v = 0; v < 8; ++v) {
        const int row = rt * 16 + v + 8 * half;
        const size_t o = (size_t)row * HH + r16;
#pragma unroll
        for (int ct = 0; ct < 4; ++ct) {
            const size_t oc = o + ct * 16;
            float s = 0.5f * (agg[oc] + x0[oc]);
            float cv = (ct == 0) ? c0[v] : (ct == 1) ? c1[v] : (ct == 2) ? c2[v] : c3[v];
            float out = ob * s + beta * cv;
            h[oc] = elu1(out) + out;
        }
    }
}

// ---------------- output GEMM: d_out = h @ w_out + b_out ----------------
__global__ __launch_bounds__(256) void k_gemm_out(const float* __restrict__ h,
                                                  const float* __restrict__ w_out,
                                                  const float* __restrict__ b_out,
                                                  float* __restrict__ out) {
    __shared__ float lwr[HH * CC];
    __shared__ float lw[HH * CC];
    if (threadIdx.x < 32) tdm_load_w(w_out, lwr, HH * CC);
    __syncthreads();
    for (int i = threadIdx.x; i < HH * CC; i += 256) {
        int k = i >> 6, n = i & 63;
        lw[(k >> 1) * 128 + 2 * n + (k & 1)] = lwr[i];
    }
    __syncthreads();

    const int wave = threadIdx.x >> 5;
    const int lane = threadIdx.x & 31;
    const int rt = blockIdx.x * 8 + wave;
    if (rt >= NN / 16) return;
    const int r16  = lane & 15;
    const int half = lane >> 4;
    const int kp   = half << 1;

    const float* hrow = h + (size_t)(rt * 16 + r16) * HH;
    v8f c0 = {}, c1 = {}, c2 = {}, c3 = {};
    for (int k0 = 0; k0 < HH; k0 += 4) {
        v2f a = *(const v2f*)(hrow + k0 + kp);
        const float* wb = lw + ((k0 + kp) >> 1) * 128 + 2 * r16;
        v2f b0 = *(const v2f*)(wb + 0);
        v2f b1 = *(const v2f*)(wb + 32);
        v2f b2 = *(const v2f*)(wb + 64);
        v2f b3 = *(const v2f*)(wb + 96);
        c0 = __builtin_amdgcn_wmma_f32_16x16x4_f32(false, a, false, b0, (short)0, c0, false, false);
        c1 = __builtin_amdgcn_wmma_f32_16x16x4_f32(false, a, false, b1, (short)0, c1, false, false);
        c2 = __builtin_amdgcn_wmma_f32_16x16x4_f32(false, a, false, b2, (short)0, c2, false, false);
        c3 = __builtin_amdgcn_wmma_f32_16x16x4_f32(false, a, false, b3, (short)0, c3, false, false);
    }

    const float bo0 = b_out[r16], bo1 = b_out[16 + r16], bo2 = b_out[32 + r16], bo3 = b_out[48 + r16];
#pragma unroll
    for (int v = 0; v < 8; ++v) {
        const int row = rt * 16 + v + 8 * half;
        const size_t o = (size_t)row * CC + r16;
        out[o +  0] = c0[v] + bo0;
        out[o + 16] = c1[v] + bo1;
        out[o + 32] = c2[v] + bo2;
        out[o + 48] = c3[v] + bo3;
    }
}

extern "C" void kernel_launch(void* const* d_in, const int* in_sizes, int n_in,
                              void* d_out, int out_size, void* d_ws, size_t ws_size,
                              hipStream_t stream) {
    const float* x        = (const float*)d_in[0];
    const int*   ei       = (const int*)d_in[1];
    const float* w_in     = (const float*)d_in[2];
    const float* b_in     = (const float*)d_in[3];
    const float* w_layers = (const float*)d_in[4];
    const float* w_out    = (const float*)d_in[5];
    const float* b_out    = (const float*)d_in[6];
    const int* src = ei;            // edge_index[0]
    const int* dst = ei + EE;       // edge_index[1]

    // workspace layout (256B-aligned offsets)
    char* ws = (char*)d_ws;
    unsigned* deg = (unsigned*)(ws + 0);        // N u32
    float* dinv   = (float*)(ws + 400128);      // N f32
    float* h      = (float*)(ws + 800256);      // N*64 f32
    float* x0     = (float*)(ws + 26400256);    // N*64 f32
    float* agg    = (float*)(ws + 52000256);    // N*64 f32

    k_zero_deg <<<(NN + 255) / 256, 256, 0, stream>>>(deg);
    k_count_deg<<<(EE + 255) / 256, 256, 0, stream>>>(dst, deg);
    k_dinv     <<<(NN + 255) / 256, 256, 0, stream>>>(deg, dinv);

    const int gemmBlocks = (NN / 16 + 7) / 8;  // 782
    k_gemm_in<<<gemmBlocks, 256, 0, stream>>>(x, w_in, b_in, h, x0);

    for (int l = 0; l < LL; ++l) {
        float beta = logf(1.0f / (float)(l + 1) + 1.0f);
        k_init_agg<<<(NN * HH + 255) / 256, 256, 0, stream>>>(dinv, h, agg);
        k_scatter<<<(int)(((long long)EE * 16 + 255) / 256), 256, 0, stream>>>(src, dst, dinv, h, agg);
        k_gemm_layer<<<gemmBlocks, 256, 0, stream>>>(agg, x0, w_layers + (size_t)l * HH * HH, beta, h);
    }

    k_gemm_out<<<gemmBlocks, 256, 0, stream>>>(h, w_out, b_out, (float*)d_out);
}